// CenterLoss_48103633715690
// MI455X (gfx1250) — compile-verified
//
#include <hip/hip_runtime.h>
#include <math.h>

#define NUM_CLASSES 1000
#define FEAT_DIM    256
#define BATCH       262144
#define ALPHA       0.5f

typedef float v2f __attribute__((ext_vector_type(2)));
typedef float v8f __attribute__((ext_vector_type(8)));

// ---------------- workspace layout (floats) ----------------
// sums        : [0, 256000)
// counts      : [256000, 257024)   (1000 used, padded)
// centers_new : [257024, 513024)
// partials    : [513024, 517120)   (4096 blocks)
#define WS_SUMS     0
#define WS_COUNTS   256000
#define WS_CNEW     257024
#define WS_PART     513024
#define ZERO_COUNT  257024   // sums + counts region

// ---------------- kernel 1: zero sums/counts ----------------
__global__ void cl_zero_kernel(float* __restrict__ p, int n) {
    int i = blockIdx.x * blockDim.x + threadIdx.x;
    if (i < n) p[i] = 0.0f;
}

// ---------------- kernel 2: segment sums via L2 atomics ----------------
// one thread per float4 of x; sums array (1MB) stays resident in the 192MB L2,
// unsafeAtomicAdd -> global_atomic_add_f32 (no-return, STOREcnt path)
__global__ __launch_bounds__(256)
void cl_segsum_kernel(const float* __restrict__ x, const int* __restrict__ labels,
                      float* __restrict__ sums, float* __restrict__ counts) {
    int i = blockIdx.x * blockDim.x + threadIdx.x;   // i in [0, BATCH*64)
    int row = i >> 6;
    int c4  = i & 63;
    int lab = labels[row];
    float4 v = ((const float4*)x)[i];
    float* base = sums + lab * FEAT_DIM + c4 * 4;
    unsafeAtomicAdd(base + 0, v.x);
    unsafeAtomicAdd(base + 1, v.y);
    unsafeAtomicAdd(base + 2, v.z);
    unsafeAtomicAdd(base + 3, v.w);
    if (c4 == 0) unsafeAtomicAdd(counts + lab, 1.0f);
}

// ---------------- kernel 3: means / keep-old-center ----------------
__global__ void cl_centers_kernel(const float* __restrict__ sums,
                                  const float* __restrict__ counts,
                                  const float* __restrict__ centers,
                                  float* __restrict__ centers_new) {
    int i = blockIdx.x * blockDim.x + threadIdx.x;   // [0, C*D)
    if (i >= NUM_CLASSES * FEAT_DIM) return;
    int c = i >> 8;
    float cnt = counts[c];
    centers_new[i] = (cnt > 0.0f) ? (sums[i] / cnt) : centers[i];
}

// ---------------- kernel 4: WMMA row-norm loss ----------------
// 128 threads = 4 waves; 64 rows/block; diff staged through LDS in two
// 128-column phases; each wave computes 16 row norms via the Gram-diagonal
// of V_WMMA_F32_16X16X4_F32 with B = A^T (identical register layout).
#define LROWS   64
#define LSTRIDE 132   // 128 cols + 4 pad floats -> conflict-free banks

__global__ __launch_bounds__(128)
void cl_loss_kernel(const float* __restrict__ x, const int* __restrict__ labels,
                    const float* __restrict__ centers_new,
                    float* __restrict__ partials) {
    __shared__ float tile[LROWS * LSTRIDE];
    __shared__ int   labs[LROWS];
    __shared__ float wsum[4];

    const int tid = threadIdx.x;
    const int rowBase = blockIdx.x * LROWS;

    if (tid < LROWS) labs[tid] = labels[rowBase + tid];
    __syncthreads();

    const int lane  = tid & 31;
    const int wave  = tid >> 5;
    const int m     = lane & 15;          // row within 16-row tile
    const int khalf = lane >> 4;          // K pair selector (0 -> K0,K1 ; 1 -> K2,K3)
    const float* rowp = &tile[(wave * 16 + m) * LSTRIDE + khalf * 2];

    v8f acc = {};
    #pragma unroll
    for (int ph = 0; ph < 2; ++ph) {
        // stage diff for columns [ph*128, ph*128+128) : 64 rows x 32 float4
        for (int i = tid; i < LROWS * 32; i += 128) {
            int r  = i >> 5;
            int c4 = i & 31;
            const float4 xv = ((const float4*)(x + (rowBase + r) * FEAT_DIM + ph * 128))[c4];
            const float4 cv = ((const float4*)(centers_new + labs[r] * FEAT_DIM + ph * 128))[c4];
            float4 dv;
            dv.x = xv.x - cv.x; dv.y = xv.y - cv.y;
            dv.z = xv.z - cv.z; dv.w = xv.w - cv.w;
            *(float4*)&tile[r * LSTRIDE + c4 * 4] = dv;
        }
        __syncthreads();
        // 32 K-chunks of 4: acc += A * A^T  (diagonal = sum of squares)
        #pragma unroll
        for (int kc = 0; kc < 32; ++kc) {
            v2f a = *(const v2f*)(rowp + kc * 4);
            acc = __builtin_amdgcn_wmma_f32_16x16x4_f32(
                false, a, false, a, (short)0, acc, false, false);
        }
        __syncthreads();
    }

    // diagonal: rows 0-7 -> lanes 0-7 (vgpr=lane), rows 8-15 -> lanes 24-31 (vgpr=lane-24)
    int idx = lane & 7;
    float g = acc[0];
    #pragma unroll
    for (int j = 1; j < 8; ++j) g = (idx == j) ? acc[j] : g;
    bool active = (lane < 8) || (lane >= 24);
    float v = active ? sqrtf(g) : 0.0f;

    #pragma unroll
    for (int off = 16; off > 0; off >>= 1) v += __shfl_xor(v, off, 32);
    if (lane == 0) wsum[wave] = v;
    __syncthreads();
    if (tid == 0) partials[blockIdx.x] = wsum[0] + wsum[1] + wsum[2] + wsum[3];
}

// ---------------- kernel 5: final deterministic reduce ----------------
__global__ __launch_bounds__(256)
void cl_reduce_kernel(const float* __restrict__ partials, int n, float* __restrict__ out) {
    __shared__ float s[256];
    float v = 0.0f;
    for (int i = threadIdx.x; i < n; i += 256) v += partials[i];
    s[threadIdx.x] = v;
    __syncthreads();
    for (int off = 128; off > 0; off >>= 1) {
        if (threadIdx.x < off) s[threadIdx.x] += s[threadIdx.x + off];
        __syncthreads();
    }
    if (threadIdx.x == 0) out[0] = ALPHA * s[0] / (float)BATCH;
}

extern "C" void kernel_launch(void* const* d_in, const int* in_sizes, int n_in,
                              void* d_out, int out_size, void* d_ws, size_t ws_size,
                              hipStream_t stream) {
    const float* x       = (const float*)d_in[0];
    const int*   labels  = (const int*)d_in[1];
    const float* centers = (const float*)d_in[2];
    float* out = (float*)d_out;

    float* ws          = (float*)d_ws;
    float* sums        = ws + WS_SUMS;
    float* counts      = ws + WS_COUNTS;
    float* centers_new = ws + WS_CNEW;
    float* partials    = ws + WS_PART;

    // 1) zero accumulators
    cl_zero_kernel<<<(ZERO_COUNT + 255) / 256, 256, 0, stream>>>(ws, ZERO_COUNT);

    // 2) segment sums (BATCH*64 float4 lanes)
    cl_segsum_kernel<<<(BATCH * 64) / 256, 256, 0, stream>>>(x, labels, sums, counts);

    // 3) centers_new
    cl_centers_kernel<<<(NUM_CLASSES * FEAT_DIM + 255) / 256, 256, 0, stream>>>(
        sums, counts, centers, centers_new);

    // 4) WMMA loss partials: 64 rows/block
    const int lossBlocks = BATCH / LROWS;   // 4096
    cl_loss_kernel<<<lossBlocks, 128, 0, stream>>>(x, labels, centers_new, partials);

    // 5) final scalar
    cl_reduce_kernel<<<1, 256, 0, stream>>>(partials, lossBlocks, out);
}